// BinarizeLayer_618475291089
// MI455X (gfx1250) — compile-verified
//
#include <hip/hip_runtime.h>
#include <hip/hip_bf16.h>

typedef __attribute__((ext_vector_type(2))) float v2f;
typedef __attribute__((ext_vector_type(4))) float vf4;
typedef __attribute__((ext_vector_type(8))) float v8f;

#define NBINS 256

// ---- ordered-uint transform for float atomic min/max ----
__device__ __forceinline__ unsigned int f2ord(float f) {
  unsigned int u = __float_as_uint(f);
  return (u & 0x80000000u) ? ~u : (u | 0x80000000u);
}
__device__ __forceinline__ float ord2f(unsigned int u) {
  return __uint_as_float((u & 0x80000000u) ? (u & 0x7fffffffu) : ~u);
}

// ws layout (uint view): [0]=min_ord  [1]=max_ord  [2..257]=hist  [258]=thresh bits

__global__ void otsu_init_kernel(unsigned int* __restrict__ ws) {
  int t = threadIdx.x;
  if (t < NBINS) ws[2 + t] = 0u;
  if (t == 0) { ws[0] = 0xFFFFFFFFu; ws[1] = 0u; }
}

__global__ void otsu_minmax_kernel(const float* __restrict__ x, long long n,
                                   unsigned int* __restrict__ ws) {
  long long n4 = n >> 2;
  long long stride = (long long)gridDim.x * blockDim.x;
  long long i = (long long)blockIdx.x * blockDim.x + threadIdx.x;
  float lo = 3.4e38f, hi = -3.4e38f;
  const vf4* x4 = (const vf4*)x;
  for (long long k = i; k < n4; k += stride) {
    vf4 v = x4[k];
    lo = fminf(lo, fminf(fminf(v.x, v.y), fminf(v.z, v.w)));
    hi = fmaxf(hi, fmaxf(fmaxf(v.x, v.y), fmaxf(v.z, v.w)));
  }
  for (long long k = (n4 << 2) + i; k < n; k += stride) {
    float v = x[k];
    lo = fminf(lo, v); hi = fmaxf(hi, v);
  }
  // wave32 reduction
  for (int off = 16; off > 0; off >>= 1) {
    lo = fminf(lo, __shfl_down(lo, off, 32));
    hi = fmaxf(hi, __shfl_down(hi, off, 32));
  }
  __shared__ float slo[8], shi[8];
  int wave = threadIdx.x >> 5;
  if ((threadIdx.x & 31) == 0) { slo[wave] = lo; shi[wave] = hi; }
  __syncthreads();
  if (threadIdx.x == 0) {
    int nw = (blockDim.x + 31) >> 5;
    for (int w = 1; w < nw; ++w) { lo = fminf(lo, slo[w]); hi = fmaxf(hi, shi[w]); }
    atomicMin(&ws[0], f2ord(lo));
    atomicMax(&ws[1], f2ord(hi));
  }
}

__global__ void otsu_hist_kernel(const float* __restrict__ x, long long n,
                                 unsigned int* __restrict__ ws) {
  __shared__ unsigned int lh[NBINS];
  for (int t = threadIdx.x; t < NBINS; t += blockDim.x) lh[t] = 0u;
  __syncthreads();
  float lo = ord2f(ws[0]);
  float hi = ord2f(ws[1]);
  float range = hi - lo;
  float scale = (range > 0.f) ? (256.0f / range) : 0.f;
  long long n4 = n >> 2;
  long long stride = (long long)gridDim.x * blockDim.x;
  long long i = (long long)blockIdx.x * blockDim.x + threadIdx.x;
  const vf4* x4 = (const vf4*)x;
  for (long long k = i; k < n4; k += stride) {
    vf4 v = x4[k];
    int b0 = min(max((int)((v.x - lo) * scale), 0), 255);
    int b1 = min(max((int)((v.y - lo) * scale), 0), 255);
    int b2 = min(max((int)((v.z - lo) * scale), 0), 255);
    int b3 = min(max((int)((v.w - lo) * scale), 0), 255);
    atomicAdd(&lh[b0], 1u);
    atomicAdd(&lh[b1], 1u);
    atomicAdd(&lh[b2], 1u);
    atomicAdd(&lh[b3], 1u);
  }
  for (long long k = (n4 << 2) + i; k < n; k += stride) {
    int b = min(max((int)((x[k] - lo) * scale), 0), 255);
    atomicAdd(&lh[b], 1u);
  }
  __syncthreads();
  for (int t = threadIdx.x; t < NBINS; t += blockDim.x)
    if (lh[t]) atomicAdd(&ws[2 + t], lh[t]);
}

// Single wave32. Inclusive cumsums of counts (c) and counts*centers (m) over
// 256 bins, computed as triangular matmuls with v_wmma_f32_16x16x4_f32:
//   view 256 values as 16x16 X (row-major); within-row prefix = X * U
//   (U upper-triangular-incl-diagonal ones), done as 4 chained K=4 WMMAs;
//   row offsets added from a 16-entry exclusive scan of row sums.
__global__ void otsu_solve_kernel(unsigned int* __restrict__ ws) {
  __shared__ float lds_c[256], lds_m[256], lds_w1[256], lds_cs[256];
  __shared__ float lds_r[32], lds_exc[32];
  int l = threadIdx.x;  // 0..31, EXEC all ones (required for WMMA)
  float lo = ord2f(ws[0]);
  float hi = ord2f(ws[1]);
  float dbin = (hi - lo) * (1.0f / 256.0f);

  for (int i = l; i < 256; i += 32) {
    float c = (float)ws[2 + i];
    float center = lo + ((float)i + 0.5f) * dbin;  // == 0.5*(edge_i + edge_{i+1})
    lds_c[i] = c;
    lds_m[i] = c * center;
  }
  __syncthreads();

  // row sums: lanes 0-15 -> rows of c, lanes 16-31 -> rows of m
  {
    const float* src = (l < 16) ? lds_c : lds_m;
    int r = l & 15;
    float s = 0.f;
    for (int cc = 0; cc < 16; ++cc) s += src[16 * r + cc];
    lds_r[l] = s;
  }
  __syncthreads();
  // exclusive prefix over the 16 row sums (separately for c and m halves)
  {
    int base = l & 16;
    int r = l & 15;
    float e = 0.f;
    for (int rp = 0; rp < r; ++rp) e += lds_r[base + rp];
    lds_exc[l] = e;
  }
  __syncthreads();

  // Y = X * U via 4 chained 16x16x4 f32 WMMAs per matrix.
  v8f dc = {0.f, 0.f, 0.f, 0.f, 0.f, 0.f, 0.f, 0.f};
  v8f dm = {0.f, 0.f, 0.f, 0.f, 0.f, 0.f, 0.f, 0.f};
  int r16 = l & 15;
  int koff = (l < 16) ? 0 : 2;
#pragma unroll
  for (int kk = 0; kk < 4; ++kk) {
    int k0 = 4 * kk + koff, k1 = k0 + 1;
    v2f bu;  // B = U slice (rows k0..), N = lane%16 : U[k][n] = (k<=n)
    bu.x = (k0 <= r16) ? 1.f : 0.f;
    bu.y = (k1 <= r16) ? 1.f : 0.f;
    v2f ac, am;  // A = X slice: row r16, cols k0,k1
    ac.x = lds_c[16 * r16 + k0]; ac.y = lds_c[16 * r16 + k1];
    am.x = lds_m[16 * r16 + k0]; am.y = lds_m[16 * r16 + k1];
    dc = __builtin_amdgcn_wmma_f32_16x16x4_f32(false, ac, false, bu, (short)0,
                                               dc, false, false);
    dm = __builtin_amdgcn_wmma_f32_16x16x4_f32(false, am, false, bu, (short)0,
                                               dm, false, false);
  }

  // add row offsets -> inclusive cumsums, write back to LDS
#pragma unroll
  for (int v = 0; v < 8; ++v) {
    int row = v + ((l < 16) ? 0 : 8);
    int idx = row * 16 + r16;
    lds_w1[idx] = dc[v] + lds_exc[row];        // cumsum(counts)
    lds_cs[idx] = dm[v] + lds_exc[16 + row];   // cumsum(counts*centers)
  }
  __syncthreads();

  float total_c = lds_exc[15] + lds_r[15];
  float total_m = lds_exc[31] + lds_r[31];

  // between-class variance at each split, argmax with first-index tie-break
  float bestv = -1.0f;
  int bestt = 0;
  for (int t = l; t < 255; t += 32) {
    float w1 = lds_w1[t];
    float w2 = total_c - lds_w1[t + 1] + lds_c[t + 1];   // reverse cumsum at t+1
    float m1 = lds_cs[t] / fmaxf(w1, 1.0f);
    float csr = total_m - lds_cs[t + 1] + lds_m[t + 1];
    float m2 = csr / fmaxf(w2, 1.0f);
    float dmm = m1 - m2;
    float var = w1 * w2 * dmm * dmm;
    if (var > bestv) { bestv = var; bestt = t; }
  }
  for (int off = 16; off > 0; off >>= 1) {
    float ov = __shfl_down(bestv, off, 32);
    int   ot = __shfl_down(bestt, off, 32);
    if (ov > bestv || (ov == bestv && ot < bestt)) { bestv = ov; bestt = ot; }
  }
  if (l == 0) {
    float thresh = lo + ((float)bestt + 0.5f) * dbin;  // centers[idx]
    ((float*)ws)[258] = thresh;
  }
}

__global__ void otsu_binarize_kernel(const float* __restrict__ x,
                                     float* __restrict__ out, long long n,
                                     const unsigned int* __restrict__ ws) {
  float thresh = ((const float*)ws)[258];
  long long n4 = n >> 2;
  long long stride = (long long)gridDim.x * blockDim.x;
  long long i = (long long)blockIdx.x * blockDim.x + threadIdx.x;
  const vf4* x4 = (const vf4*)x;
  vf4* o4 = (vf4*)out;
  for (long long k = i; k < n4; k += stride) {
    vf4 v = x4[k];
    vf4 r;
    r.x = (v.x > thresh) ? 1.f : 0.f;
    r.y = (v.y > thresh) ? 1.f : 0.f;
    r.z = (v.z > thresh) ? 1.f : 0.f;
    r.w = (v.w > thresh) ? 1.f : 0.f;
    // NT store: input (128MB) fits in the 192MB L2; keep it resident by not
    // letting the output stream evict it.
    __builtin_nontemporal_store(r, &o4[k]);
  }
  for (long long k = (n4 << 2) + i; k < n; k += stride)
    out[k] = (x[k] > thresh) ? 1.f : 0.f;
}

extern "C" void kernel_launch(void* const* d_in, const int* in_sizes, int n_in,
                              void* d_out, int out_size, void* d_ws, size_t ws_size,
                              hipStream_t stream) {
  const float* x = (const float*)d_in[0];
  float* out = (float*)d_out;
  long long n = (long long)in_sizes[0];
  unsigned int* ws = (unsigned int*)d_ws;

  const int threads = 256;   // 8 wave32 per workgroup
  const int blocks = 2048;

  otsu_init_kernel<<<1, 256, 0, stream>>>(ws);
  otsu_minmax_kernel<<<blocks, threads, 0, stream>>>(x, n, ws);
  otsu_hist_kernel<<<blocks, threads, 0, stream>>>(x, n, ws);
  otsu_solve_kernel<<<1, 32, 0, stream>>>(ws);
  otsu_binarize_kernel<<<blocks, threads, 0, stream>>>(x, out, (long long)out_size, ws);
}